// LeiNet_6064493822350
// MI455X (gfx1250) — compile-verified
//
#include <hip/hip_runtime.h>

typedef __attribute__((ext_vector_type(2))) float v2f;
typedef __attribute__((ext_vector_type(4))) float v4f;
typedef __attribute__((ext_vector_type(8))) float v8f;

#define N_NEURON 128
#define N_RULES  512
#define BATCH    4096
#define BTILE    16
#define LDS_STRIDE 132   // pad vs 128 to avoid bank conflicts (132 % 64 = 4)

// One block = 256 threads = 8 wave32s. Block owns a 16-batch tile and all 128
// neurons: wave w computes neurons [16w, 16w+16) via V_WMMA_F32_16X16X4_F32,
// writes head tile to LDS, then 16 lanes do the foot GEMV + softmax.
__global__ __launch_bounds__(256)
void leinet_fused_kernel(const float* __restrict__ x,
                         const float* __restrict__ head_w,
                         const float* __restrict__ head_b,
                         const float* __restrict__ foot_w,
                         const float* __restrict__ foot_b,
                         float* __restrict__ out)
{
    __shared__ float lds_head[BTILE * LDS_STRIDE];

    const int tid  = threadIdx.x;
    const int wave = tid >> 5;        // 0..7
    const int lane = tid & 31;
    const int m    = lane & 15;       // A-matrix row within tile
    const int kh   = lane >> 4;       // K half: lanes 0-15 -> K 0..1, 16-31 -> K 2..3
    const int b0   = blockIdx.x * BTILE;

    for (int t = 0; t < 16; ++t) {
        const int n = wave * 16 + t;
        // Per-lane stream base: row (b0+m) of x[n,:,:], K offset 4*kh floats.
        const float* xrow = x + ((size_t)n * BATCH + (size_t)(b0 + m)) * N_RULES + 4 * kh;
        const float* wrow = head_w + (size_t)n * N_RULES + 4 * kh;

        // Warm GL2 for the next neuron's rows while we chew on this one.
        if (t < 15) {
            __builtin_prefetch(xrow + (size_t)BATCH * N_RULES, 0, 1);
        }

        v8f acc = {};
        #pragma unroll 4
        for (int r0 = 0; r0 < N_RULES; r0 += 8) {
            // lo lanes: x[.., r0..r0+3]; hi lanes: x[.., r0+4..r0+7]
            v4f a4 = *(const v4f*)(xrow + r0);
            v4f b4 = *(const v4f*)(wrow + r0);
            v2f a0 = { a4.x, a4.y };  v2f bb0 = { b4.x, b4.y };
            v2f a1 = { a4.z, a4.w };  v2f bb1 = { b4.z, b4.w };
            // WMMA0 covers K = {r0, r0+1, r0+4, r0+5}; WMMA1 the other four.
            // A and B use identical K->slot maps, so the accumulation is exact.
            acc = __builtin_amdgcn_wmma_f32_16x16x4_f32(
                      false, a0, false, bb0, (short)0, acc, false, false);
            acc = __builtin_amdgcn_wmma_f32_16x16x4_f32(
                      false, a1, false, bb1, (short)0, acc, false, false);
        }

        // D layout: lane l, VGPR v -> row M = v + 8*(l>=16), col l&15 (all cols
        // equal by construction). Lanes 0 and 16 carry the 16 distinct values.
        if (m == 0) {
            #pragma unroll
            for (int v = 0; v < 8; ++v)
                lds_head[(v + 8 * kh) * LDS_STRIDE + n] = acc[v];
        }
    }
    __syncthreads();

    // Foot stage: logits = head @ foot_w^T + foot_b, softmax over 2 classes.
    if (tid < BTILE) {
        float l0 = foot_b[0];
        float l1 = foot_b[1];
        #pragma unroll 4
        for (int n = 0; n < N_NEURON; ++n) {
            float h = lds_head[tid * LDS_STRIDE + n] + head_b[n];
            l0 = fmaf(h, foot_w[n], l0);
            l1 = fmaf(h, foot_w[N_NEURON + n], l1);
        }
        float mx = fmaxf(l0, l1);
        float e0 = __expf(l0 - mx);
        float e1 = __expf(l1 - mx);
        float inv = 1.0f / (e0 + e1);
        out[(size_t)(b0 + tid) * 2 + 0] = e0 * inv;
        out[(size_t)(b0 + tid) * 2 + 1] = e1 * inv;
    }
}

extern "C" void kernel_launch(void* const* d_in, const int* in_sizes, int n_in,
                              void* d_out, int out_size, void* d_ws, size_t ws_size,
                              hipStream_t stream) {
    (void)in_sizes; (void)n_in; (void)out_size; (void)d_ws; (void)ws_size;
    const float* x      = (const float*)d_in[0];
    const float* head_w = (const float*)d_in[1];
    const float* head_b = (const float*)d_in[2];
    const float* foot_w = (const float*)d_in[3];
    const float* foot_b = (const float*)d_in[4];
    float* out = (float*)d_out;

    dim3 grid(BATCH / BTILE);   // 256 blocks
    dim3 block(256);            // 8 wave32s
    leinet_fused_kernel<<<grid, block, 0, stream>>>(x, head_w, head_b,
                                                    foot_w, foot_b, out);
}